// prune_voxel_81329500717056
// MI455X (gfx1250) — compile-verified
//
#include <hip/hip_runtime.h>
#include <stdint.h>

#define BATCH 8

// ---------------------------------------------------------------------------
// CDNA5 feature detection (compile-time, guarded so host pass / older clang
// still compiles cleanly).
// ---------------------------------------------------------------------------
#if defined(__HIP_DEVICE_COMPILE__) && defined(__gfx1250__) && \
    __has_builtin(__builtin_amdgcn_global_load_async_to_lds_b128)
#define USE_ASYNC 1
#else
#define USE_ASYNC 0
#endif

#if USE_ASYNC
// Builtin signature (from clang diagnostic): params are pointers to GCC-vector
// int4 in AS1 (global) and AS3 (LDS).
typedef int v4i __attribute__((vector_size(16)));
typedef __attribute__((address_space(1))) v4i v4i_as1;
typedef __attribute__((address_space(3))) v4i v4i_as3;
#endif

__device__ __forceinline__ unsigned ballot32(bool p) {
#if defined(__HIP_DEVICE_COMPILE__)
#if __has_builtin(__builtin_amdgcn_ballot_w32)
  return __builtin_amdgcn_ballot_w32(p);
#else
  return (unsigned)__ballot(p);
#endif
#else
  (void)p;
  return 0u;
#endif
}

template <int N>
__device__ __forceinline__ void wait_asynccnt() {
#if defined(__HIP_DEVICE_COMPILE__) && defined(__gfx1250__)
#if __has_builtin(__builtin_amdgcn_s_wait_asynccnt)
  __builtin_amdgcn_s_wait_asynccnt(N);
#else
  asm volatile("s_wait_asynccnt %0" ::"i"(N) : "memory");
#endif
#endif
}

// ---------------------------------------------------------------------------
// Feature gather (C==64 fast path).  rowIdx[] is the precompacted map from
// output row -> source feature row (no division in the hot loop; all 32-bit).
// Each wave moves 512 B per chunk (2 output rows, 16 lanes x float4 per row),
// staged through LDS with GLOBAL_LOAD_ASYNC_TO_LDS_B128 (per-lane gather
// addresses, ASYNCcnt), double-buffered: issue chunk c+stride,
// s_wait_asynccnt<=1 (async loads retire in order) so chunk c is resident,
// then ds_read + global_store_b128.
// ---------------------------------------------------------------------------
__global__ __launch_bounds__(256) void gather_features64(
    const float* __restrict__ feat, const unsigned* __restrict__ rowIdx,
    float* __restrict__ out, unsigned nChunks) {
  const unsigned lane = threadIdx.x & 31u;
  const unsigned w = threadIdx.x >> 5;
  const unsigned half = lane >> 4;   // 0: row 2c, 1: row 2c+1
  const unsigned comp = lane & 15u;  // float4 slot within the 64-float row
  unsigned gw = blockIdx.x * (blockDim.x >> 5) + w;
  const unsigned stride = gridDim.x * (blockDim.x >> 5);

  auto srcPtr = [&](unsigned c) -> const float* {
    unsigned row = (c << 1) + half;
    unsigned fi = rowIdx[row];
    return feat + (size_t)fi * 64u + (size_t)comp * 4u;
  };

#if USE_ASYNC
  __shared__ float4 stage[8][2][32];  // 8 waves x 2 buffers x 512B
  if (gw >= nChunks) return;
  {
    const float* s = srcPtr(gw);
    __builtin_amdgcn_global_load_async_to_lds_b128(
        (v4i_as1*)s, (v4i_as3*)&stage[w][0][lane], 0, 0);
  }
  unsigned c = gw;
  int p = 0;
  for (;;) {
    unsigned cn = c + stride;
    const bool more = cn < nChunks;
    if (more) {
      const float* s = srcPtr(cn);
      __builtin_amdgcn_global_load_async_to_lds_b128(
          (v4i_as1*)s, (v4i_as3*)&stage[w][p ^ 1][lane], 0, 0);
      wait_asynccnt<1>();  // loads retire in order -> chunk c is in LDS
    } else {
      wait_asynccnt<0>();
    }
    float4 v = stage[w][p][lane];
    unsigned row = (c << 1) + half;
    *(float4*)(out + (size_t)row * 64u + (size_t)comp * 4u) = v;
    if (!more) break;
    c = cn;
    p ^= 1;
  }
#else
  for (unsigned c = gw; c < nChunks; c += stride) {
    const float* s = srcPtr(c);
#if defined(__HIP_DEVICE_COMPILE__)
    __builtin_prefetch(s, 0, 0);  // global_prefetch_b8 on gfx1250
#endif
    float4 v = *(const float4*)s;
    unsigned row = (c << 1) + half;
    *(float4*)(out + (size_t)row * 64u + (size_t)comp * 4u) = v;
  }
#endif
}

// ---------------------------------------------------------------------------
// Build sort keys.  key = ~monotonic(float) so that ASCENDING radix order
// == DESCENDING float order; stable sort then reproduces jax.lax.top_k
// ordering (ties -> lower original index first).  Payload .y = global row.
// ---------------------------------------------------------------------------
__global__ void build_keys(const float* __restrict__ feat, uint2* __restrict__ keys,
                           int C, long long N) {
  long long i = (long long)blockIdx.x * blockDim.x + threadIdx.x;
  if (i >= N) return;
  float f = feat[(size_t)i * C];  // score = feature channel 0
  unsigned u = __float_as_uint(f);
  u = (u & 0x80000000u) ? ~u : (u | 0x80000000u);  // ascending-order map
  keys[i] = make_uint2(~u, (unsigned)i);           // invert -> descending
}

// ---------------------------------------------------------------------------
// Stable LSD radix sort, 8-bit digits, one workgroup (1024 thr = 32 waves)
// per batch segment of nb elements.  Stable scatter rank =
// runningBase[digit] + exclusive-wave-prefix[digit] + in-wave rank, where
// the in-wave rank comes from wave32 ballot multi-split.
// ---------------------------------------------------------------------------
__global__ __launch_bounds__(1024) void radix_pass(const uint2* __restrict__ src,
                                                   uint2* __restrict__ dst,
                                                   int nb, int shift) {
  __shared__ unsigned sBase[256];       // histogram -> running exclusive base
  __shared__ unsigned sWave[32 * 256];  // per-wave digit counts -> prefixes
  __shared__ unsigned sTot[256];        // per-chunk digit totals
  const int tid = threadIdx.x;
  const size_t base = (size_t)blockIdx.x * (size_t)nb;

  if (tid < 256) sBase[tid] = 0u;
  __syncthreads();
  for (int i = tid; i < nb; i += 1024) {
    unsigned d = (src[base + i].x >> shift) & 255u;
    atomicAdd(&sBase[d], 1u);
  }
  __syncthreads();
  if (tid == 0) {  // exclusive scan of 256 bins (cheap, once per pass)
    unsigned run = 0;
    for (int j = 0; j < 256; ++j) {
      unsigned t = sBase[j];
      sBase[j] = run;
      run += t;
    }
  }
  __syncthreads();

  const int lane = tid & 31;
  const int w = tid >> 5;
  const int nchunks = (nb + 1023) >> 10;
  for (int c = 0; c < nchunks; ++c) {
    for (int i = tid; i < 32 * 256; i += 1024) sWave[i] = 0u;
    const int idx = (c << 10) + tid;
    const bool valid = idx < nb;
    uint2 kv = make_uint2(0u, 0u);
    unsigned d = 0u;
    if (valid) {
      kv = src[base + idx];
      d = (kv.x >> shift) & 255u;
    }
    __syncthreads();  // zeroing done

    // wave32 multi-split: lanes with equal digit form one group
    unsigned vmask = ballot32(valid);
    unsigned mask = 0xFFFFFFFFu;
#pragma unroll
    for (int bit = 0; bit < 8; ++bit) {
      unsigned vote = ballot32(valid && ((d >> bit) & 1u));
      mask &= ((d >> bit) & 1u) ? vote : ~vote;
    }
    mask &= vmask;
    unsigned localRank = 0u;
    if (valid) {
      localRank = (unsigned)__popc(mask & ((1u << lane) - 1u));
      if (localRank == 0u)  // group leader publishes group size
        sWave[(w << 8) + d] = (unsigned)__popc(mask);
    }
    __syncthreads();  // leader writes done

    if (tid < 256) {  // exclusive prefix over the 32 waves, per digit
      unsigned run = 0;
      for (int w2 = 0; w2 < 32; ++w2) {
        unsigned t = sWave[(w2 << 8) + tid];
        sWave[(w2 << 8) + tid] = run;
        run += t;
      }
      sTot[tid] = run;
    }
    __syncthreads();  // prefixes ready

    if (valid) {
      unsigned pos = sBase[d] + sWave[(w << 8) + d] + localRank;
      dst[base + pos] = kv;
    }
    __syncthreads();  // all reads of sBase done
    if (tid < 256) sBase[tid] += sTot[tid];
    __syncthreads();
  }
}

// ---------------------------------------------------------------------------
// Compact the sorted keys into a flat output-row -> feature-row map, and set
// the mask bits (fused; mask must be zeroed first).  The single 32-bit
// division by K lives here, once per output row, instead of in the gathers.
// ---------------------------------------------------------------------------
__global__ void compact_indices(const uint2* __restrict__ skeys,
                                unsigned* __restrict__ rowIdx,
                                float* __restrict__ m, int nb, int K,
                                int totalRows) {
  int g = blockIdx.x * blockDim.x + threadIdx.x;
  if (g >= totalRows) return;
  unsigned b = (unsigned)g / (unsigned)K;
  unsigned r = (unsigned)g - b * (unsigned)K;
  unsigned fi = skeys[(size_t)b * nb + r].y;  // global feature row
  rowIdx[g] = fi;
  m[fi] = 1.0f;  // fi is also the flat mask index
}

// Generic feature gather for C != 64 (not hit for the reference shapes).
__global__ void gather_features_generic(const float* __restrict__ feat,
                                        const unsigned* __restrict__ rowIdx,
                                        float* __restrict__ out, int C,
                                        long long total) {
  long long g = (long long)blockIdx.x * blockDim.x + threadIdx.x;
  if (g >= total) return;
  int c = (int)(g % C);
  long long row = g / C;
  unsigned fi = rowIdx[row];
  out[g] = feat[(size_t)fi * C + c];
}

// ---------------------------------------------------------------------------
// Coord gather (int -> float, values < 1024 so exact).
// ---------------------------------------------------------------------------
__global__ void gather_coords(const int* __restrict__ coords,
                              const unsigned* __restrict__ rowIdx,
                              float* __restrict__ out, unsigned total) {
  unsigned g = blockIdx.x * blockDim.x + threadIdx.x;
  if (g >= total) return;
  unsigned row = g / 3u;  // constant divisor -> mul-shift
  unsigned j = g - row * 3u;
  unsigned fi = rowIdx[row];
  out[g] = (float)coords[(size_t)fi * 3u + j];
}

__global__ void mask_zero(float* __restrict__ m, long long N) {
  long long i = (long long)blockIdx.x * blockDim.x + threadIdx.x;
  if (i < N) m[i] = 0.0f;
}

// ---------------------------------------------------------------------------
// Launch.  k is derived on the HOST from out_size (out_size = B*k*(C+3)+N),
// so no device read of the k scalar is needed (graph-capture safe).
// ---------------------------------------------------------------------------
extern "C" void kernel_launch(void* const* d_in, const int* in_sizes, int n_in,
                              void* d_out, int out_size, void* d_ws,
                              size_t ws_size, hipStream_t stream) {
  (void)n_in;
  (void)ws_size;
  const float* features = (const float*)d_in[0];
  const int* coords = (const int*)d_in[1];

  const long long NC = in_sizes[0];
  const long long N3 = in_sizes[1];
  const long long N = N3 / 3;          // 800000
  const int C = (int)(NC / N);         // 64
  const int nb = (int)(N / BATCH);     // 100000
  const long long K = ((long long)out_size - N) / ((long long)BATCH * (C + 3));
  const long long totalRows = (long long)BATCH * K;

  uint2* ka = (uint2*)d_ws;                 // ping (N * 8B)
  uint2* kb = ka + N;                       // pong (N * 8B)
  unsigned* rowIdx = (unsigned*)(kb + N);   // totalRows * 4B

  float* out_f = (float*)d_out;
  float* out_c = out_f + totalRows * C;
  float* out_m = out_c + totalRows * 3;

  build_keys<<<(int)((N + 255) / 256), 256, 0, stream>>>(features, ka, C, N);

  for (int pass = 0; pass < 4; ++pass) {  // ka->kb->ka->kb->ka
    const uint2* s = (pass & 1) ? kb : ka;
    uint2* d = (pass & 1) ? ka : kb;
    radix_pass<<<BATCH, 1024, 0, stream>>>(s, d, nb, pass * 8);
  }
  const uint2* skeys = ka;  // fully sorted (descending score, stable)

  mask_zero<<<(int)((N + 255) / 256), 256, 0, stream>>>(out_m, N);
  compact_indices<<<(int)((totalRows + 255) / 256), 256, 0, stream>>>(
      skeys, rowIdx, out_m, nb, (int)K, (int)totalRows);

  if (C == 64) {
    gather_features64<<<2048, 256, 0, stream>>>(features, rowIdx, out_f,
                                                (unsigned)(totalRows >> 1));
  } else {
    long long tot = totalRows * C;
    gather_features_generic<<<(int)((tot + 255) / 256), 256, 0, stream>>>(
        features, rowIdx, out_f, C, tot);
  }
  gather_coords<<<(int)((totalRows * 3 + 255) / 256), 256, 0, stream>>>(
      coords, rowIdx, out_c, (unsigned)(totalRows * 3));
}